// SusonoTopKRouter_61753039781960
// MI455X (gfx1250) — compile-verified
//
#include <hip/hip_runtime.h>
#include <hip/hip_bf16.h>

#define DDIM 4096
#define NEXP 64
#define TOPK 8
#define TOK_PER_BLK 32
#define LDS_STRIDE 68   // 64 + 4 pad: conflict-free-ish banks, 16B-aligned rows

typedef __attribute__((ext_vector_type(2))) float v2f;
typedef __attribute__((ext_vector_type(8))) float v8f;

__global__ __launch_bounds__(256)
void susono_topk_router_kernel(const float* __restrict__ x,
                               const float* __restrict__ w,
                               float* __restrict__ probs,
                               float* __restrict__ topw,
                               int*   __restrict__ topi)
{
    __shared__ float lds_logits[TOK_PER_BLK][LDS_STRIDE];

    const int tid  = threadIdx.x;
    const int wid  = tid >> 5;       // wave id 0..7
    const int lane = tid & 31;
    const int hf   = lane >> 4;      // 0: K={0,1}, 1: K={2,3}
    const int lm   = lane & 15;      // row (A) / col (B) within tile

    const int tokTile   = wid >> 2;  // 0..1  -> 16-token tile
    const int expTile   = wid & 3;   // 0..3  -> 16-expert tile
    const int tokenBase = blockIdx.x * TOK_PER_BLK;

    // A: 16x4 fp32 tile per wave. lane(lm,hf) holds x[token, k + 2*hf + {0,1}]
    const float* aRow = x + (size_t)(tokenBase + tokTile * 16 + lm) * DDIM + 2 * hf;
    // B: 4x16 fp32 tile. lane(lm,hf) holds w[expert, k + 2*hf + {0,1}]
    const float* bRow = w + (size_t)(expTile * 16 + lm) * DDIM + 2 * hf;

    v8f c0 = {};
    v8f c1 = {};

    #pragma unroll 4
    for (int k = 0; k < DDIM; k += 8) {
        v2f a0 = *(const v2f*)(aRow + k);
        v2f b0 = *(const v2f*)(bRow + k);
        v2f a1 = *(const v2f*)(aRow + k + 4);
        v2f b1 = *(const v2f*)(bRow + k + 4);
        // D = A(16x4) * B(4x16) + C, fp32 in/out (matches fp32 reference matmul)
        c0 = __builtin_amdgcn_wmma_f32_16x16x4_f32(false, a0, false, b0,
                                                   (short)0, c0, false, false);
        c1 = __builtin_amdgcn_wmma_f32_16x16x4_f32(false, a1, false, b1,
                                                   (short)0, c1, false, false);
    }
    v8f c = c0 + c1;

    // C/D layout: VGPR r -> token row (r + 8*hf), column = lm
    {
        const int rowBase = tokTile * 16 + hf * 8;
        const int col     = expTile * 16 + lm;
        #pragma unroll
        for (int r = 0; r < 8; ++r)
            lds_logits[rowBase + r][col] = c[r];
    }
    __syncthreads();

    // ---- softmax + top-8 epilogue: one thread per token (32 tokens/block) ----
    if (tid < TOK_PER_BLK) {
        float* row = lds_logits[tid];
        const size_t t = (size_t)tokenBase + tid;

        float mx = row[0];
        #pragma unroll
        for (int e = 1; e < NEXP; ++e) mx = fmaxf(mx, row[e]);

        float sum = 0.0f;
        #pragma unroll
        for (int e = 0; e < NEXP; ++e) {
            float ev = expf(row[e] - mx);
            row[e] = ev;
            sum += ev;
        }
        const float inv = 1.0f / sum;
        #pragma unroll
        for (int e = 0; e < NEXP; ++e) row[e] *= inv;

        // stream full routing_probs row: 16 x float4, contiguous 256B
        float4* __restrict__ outp = (float4*)(probs + t * NEXP);
        const float4* rp = (const float4*)row;
        #pragma unroll
        for (int q = 0; q < NEXP / 4; ++q) outp[q] = rp[q];

        // 8-pass selection top-k on LDS row (probs >= 0 so -1 masks)
        float tw[TOPK];
        int   ti[TOPK];
        float wsum = 0.0f;
        #pragma unroll
        for (int j = 0; j < TOPK; ++j) {
            float best = -1.0f;
            int   bi   = 0;
            for (int e = 0; e < NEXP; ++e) {
                float v = row[e];
                if (v > best) { best = v; bi = e; }
            }
            tw[j] = best;
            ti[j] = bi;
            wsum += best;
            row[bi] = -1.0f;
        }
        const float invw = 1.0f / (wsum + 1e-6f);
        #pragma unroll
        for (int j = 0; j < TOPK; ++j) {
            topw[t * TOPK + j] = tw[j] * invw;
            topi[t * TOPK + j] = ti[j];
        }
    }
}

extern "C" void kernel_launch(void* const* d_in, const int* in_sizes, int n_in,
                              void* d_out, int out_size, void* d_ws, size_t ws_size,
                              hipStream_t stream) {
    (void)n_in; (void)d_ws; (void)ws_size; (void)out_size;
    const float* x = (const float*)d_in[0];   // [T, 4096] f32
    const float* w = (const float*)d_in[1];   // [64, 4096] f32

    const int T = in_sizes[0] / DDIM;         // 8192

    float* probs = (float*)d_out;                          // [T, 64]
    float* topw  = probs + (size_t)T * NEXP;               // [T, 8]
    int*   topi  = (int*)(topw + (size_t)T * TOPK);        // [T, 8] (int32 bits)

    dim3 grid(T / TOK_PER_BLK);   // 256 blocks
    dim3 block(256);              // 8 waves
    susono_topk_router_kernel<<<grid, block, 0, stream>>>(x, w, probs, topw, topi);
}